// QueryAndGroup_78065325572418
// MI455X (gfx1250) — compile-verified
//
#include <hip/hip_runtime.h>

// QueryAndGroup for MI455X (gfx1250, wave32).
// Ball-query distance test via V_WMMA_F32_16X16X4_F32:
//   d2(q,p) = |q|^2 - 2 q.p + |p|^2  < r^2
//   A(16x4) row m = (-2qx, -2qy, -2qz, 1)   (lane m: K0,K1 ; lane m+16: K2,K3)
//   B(4x16) col n = ( px,  py,  pz, |p|^2)  (lane n: K0,K1 ; lane n+16: K2,K3)
//   D[v]: lanes 0-15 -> row v, col=lane ; lanes 16-31 -> row v+8, col=lane-16
// N-scan is split into SEGS segments per query tile (4x waves for latency
// hiding); per-segment in-order hit lists (truncated at 32) are merged in
// order by the gather kernel -> identical "first 32 in order" semantics.

typedef __attribute__((ext_vector_type(2))) float v2f;
typedef __attribute__((ext_vector_type(8))) float v8f;

static constexpr int   BB      = 2;
static constexpr int   NN      = 16384;
static constexpr int   NPOINT  = 4096;
static constexpr int   CC      = 64;
static constexpr int   NSAMPLE = 32;
static constexpr float RADIUS  = 0.1f;
static constexpr int   SEGS    = 4;
static constexpr int   SEGLEN  = NN / SEGS;   // 4096

// Workspace layout (bytes):
static constexpr size_t WS_P4    = 0;                                      // float4[B*N]
static constexpr size_t WS_CNTS  = WS_P4 + (size_t)BB * NN * 16;           // int[B*NPOINT*SEGS]
static constexpr size_t WS_LISTS = WS_CNTS + (size_t)BB * NPOINT * SEGS * 4; // int[B*NPOINT*SEGS*32]

// ---------------------------------------------------------------------------
// Kernel 0: pack points as (x, y, z, |p|^2) for single-b128 inner-loop loads.
// ---------------------------------------------------------------------------
__global__ void qg_prep_kernel(const float* __restrict__ xyz,
                               float4* __restrict__ p4, int total) {
    int i = blockIdx.x * blockDim.x + threadIdx.x;
    if (i < total) {
        float x = xyz[3 * i + 0];
        float y = xyz[3 * i + 1];
        float z = xyz[3 * i + 2];
        p4[i] = make_float4(x, y, z, x * x + y * y + z * z);
    }
}

// ---------------------------------------------------------------------------
// Kernel 1: ball query. One wave per (16-query tile, N-segment).
// ---------------------------------------------------------------------------
__global__ __launch_bounds__(256) void qg_ball_kernel(
    const float4* __restrict__ p4,
    const float*  __restrict__ new_xyz,
    int* __restrict__ cnts,     // [B*NPOINT][SEGS]
    int* __restrict__ lists) {  // [B*NPOINT][SEGS][NSAMPLE]

    const int lane = threadIdx.x & 31;
    const int wid  = threadIdx.x >> 5;
    const int w    = blockIdx.x * (blockDim.x >> 5) + wid;
    const int tilesPerBatch = NPOINT / 16;                  // 256
    if (w >= BB * tilesPerBatch * SEGS) return;             // wave-uniform

    const int tile = w >> 2;           // SEGS == 4
    const int seg  = w & (SEGS - 1);
    const int b    = tile / tilesPerBatch;
    const int q0   = (tile % tilesPerBatch) * 16;
    const int segStart = seg * SEGLEN;
    const int segEnd   = segStart + SEGLEN;

    // Load this lane's query (lanes m and m+16 load query q0+m).
    const int m = lane & 15;
    const int q = q0 + m;
    const float qx = new_xyz[(b * NPOINT + q) * 3 + 0];
    const float qy = new_xyz[(b * NPOINT + q) * 3 + 1];
    const float qz = new_xyz[(b * NPOINT + q) * 3 + 2];
    const float thresh = RADIUS * RADIUS - (qx * qx + qy * qy + qz * qz);

    // A matrix: row m = (-2qx, -2qy, -2qz, 1)
    v2f av;
    av[0] = (lane < 16) ? (-2.0f * qx) : (-2.0f * qz);
    av[1] = (lane < 16) ? (-2.0f * qy) : 1.0f;

    // Per-D-VGPR row thresholds: lanes 0-15 need row v, lanes 16-31 row v+8.
    float thv[8];
#pragma unroll
    for (int v = 0; v < 8; ++v)
        thv[v] = __shfl(thresh, (lane < 16) ? v : (v + 8), 32);

    // Lane-dependent mux controls for ballot->row-bits packing.
    const bool  s0 = (lane & 1) != 0;
    const bool  s1 = (lane & 2) != 0;
    const bool  s2 = (lane & 4) != 0;
    const unsigned hshift = (lane & 8) ? 16u : 0u;

    const float4* pb = p4 + b * NN;
    const v8f czero = {};

    int cnt   = (lane < 16) ? 0 : NSAMPLE;  // lanes >=16 count as done
    const int row      = b * NPOINT + q0 + lane;            // valid lanes <16
    const int listBase = (row * SEGS + seg) * NSAMPLE;

    // Software pipeline: prefetch next iteration's two b128 loads.
    float4 c0 = pb[segStart + m];
    float4 c1 = pb[segStart + 16 + m];

    int base = segStart;
    for (; base < segEnd; base += 32) {
        int nb = base + 32;
        if (nb >= segEnd) nb = segStart;    // wrap: always a valid address
        const float4 n0 = pb[nb + m];
        const float4 n1 = pb[nb + 16 + m];

        unsigned word = 0;  // lane r(<16): bit k -> point base+k hits row r
#pragma unroll
        for (int k = 0; k < 2; ++k) {
            const float4 p = k ? c1 : c0;
            v2f bv;
            bv[0] = (lane < 16) ? p.x : p.z;
            bv[1] = (lane < 16) ? p.y : p.w;
            v8f d = __builtin_amdgcn_wmma_f32_16x16x4_f32(
                false, av, false, bv, (short)0, czero, false, false);
            unsigned bal[8];
#pragma unroll
            for (int v = 0; v < 8; ++v)
                bal[v] = __builtin_amdgcn_ballot_w32(d[v] < thv[v]);
            // 8 -> 1 mux by row index (lane&7), then half-select (lane&8).
            unsigned ta  = s0 ? bal[1] : bal[0];
            unsigned tb  = s0 ? bal[3] : bal[2];
            unsigned tc  = s0 ? bal[5] : bal[4];
            unsigned td  = s0 ? bal[7] : bal[6];
            unsigned te  = s1 ? tb : ta;
            unsigned tf  = s1 ? td : tc;
            unsigned sel = s2 ? tf : te;
            word |= ((sel >> hshift) & 0xFFFFu) << (k * 16);
        }

        // In-order index extraction (bit position == point order).
        if (lane < 16) {
            while (word != 0u && cnt < NSAMPLE) {
                const int pos = __builtin_ctz(word);
                word &= word - 1u;
                lists[listBase + cnt] = base + pos;
                ++cnt;
            }
        }

        c0 = n0;
        c1 = n1;

        if (__builtin_amdgcn_ballot_w32(cnt >= NSAMPLE) == 0xFFFFFFFFu)
            break;  // all 16 rows full within this segment: uniform early exit
    }

    if (lane < 16)
        cnts[row * SEGS + seg] = cnt;
}

// ---------------------------------------------------------------------------
// Kernel 2: merge segment lists (in segment order) + gather + concat.
// One wave per (b, p); lane = sample slot.  out: (B, 3+C, NPOINT, NSAMPLE)
// ---------------------------------------------------------------------------
__global__ __launch_bounds__(256) void qg_group_kernel(
    const float* __restrict__ xyz,
    const float* __restrict__ new_xyz,
    const float* __restrict__ feats,
    const int*   __restrict__ cnts,
    const int*   __restrict__ lists,
    float* __restrict__ out) {

    const int lane = threadIdx.x & 31;
    const int wid  = threadIdx.x >> 5;
    const int pair = blockIdx.x * (blockDim.x >> 5) + wid;  // (b,p) flat = row
    if (pair >= BB * NPOINT) return;

    const int b = pair / NPOINT;
    const int p = pair - b * NPOINT;

    // Merge: lane s takes the s-th hit across segments in order; pad with the
    // first overall hit (0 if no hits).
    int j = 0;
    int firstIdx = 0;
    bool haveFirst = false;
    bool found = false;
    int rem = lane;
#pragma unroll
    for (int g = 0; g < SEGS; ++g) {
        const int cg = cnts[pair * SEGS + g];               // wave-uniform
        const int* Lg = lists + (pair * SEGS + g) * NSAMPLE;
        if (!haveFirst && cg > 0) { firstIdx = Lg[0]; haveFirst = true; }
        if (!found) {
            if (rem < cg) { j = Lg[rem]; found = true; }
            else rem -= cg;
        }
    }
    if (!found) j = firstIdx;

    const float qx = new_xyz[(b * NPOINT + p) * 3 + 0];
    const float qy = new_xyz[(b * NPOINT + p) * 3 + 1];
    const float qz = new_xyz[(b * NPOINT + p) * 3 + 2];
    const float px = xyz[(b * NN + j) * 3 + 0];
    const float py = xyz[(b * NN + j) * 3 + 1];
    const float pz = xyz[(b * NN + j) * 3 + 2];

    const size_t chanStride = (size_t)NPOINT * NSAMPLE;
    float* ob = out + ((size_t)b * (3 + CC)) * chanStride + (size_t)p * NSAMPLE + lane;

    ob[0 * chanStride] = px - qx;
    ob[1 * chanStride] = py - qy;
    ob[2 * chanStride] = pz - qz;

    const float* fb = feats + (size_t)b * CC * NN + j;
#pragma unroll 4
    for (int c = 0; c < CC; ++c)
        ob[(3 + c) * chanStride] = fb[(size_t)c * NN];
}

// ---------------------------------------------------------------------------
extern "C" void kernel_launch(void* const* d_in, const int* in_sizes, int n_in,
                              void* d_out, int out_size, void* d_ws, size_t ws_size,
                              hipStream_t stream) {
    const float* xyz     = (const float*)d_in[0];  // (B, N, 3)
    const float* new_xyz = (const float*)d_in[1];  // (B, NPOINT, 3)
    const float* feats   = (const float*)d_in[2];  // (B, C, N)
    float* out = (float*)d_out;                    // (B, 3+C, NPOINT, NSAMPLE)

    float4* p4    = (float4*)((char*)d_ws + WS_P4);
    int*    cnts  = (int*)((char*)d_ws + WS_CNTS);
    int*    lists = (int*)((char*)d_ws + WS_LISTS);

    // 0) pack points
    {
        const int total = BB * NN;
        qg_prep_kernel<<<(total + 255) / 256, 256, 0, stream>>>(xyz, p4, total);
    }
    // 1) ball query (WMMA): 512 tiles x 4 segments = 2048 waves, 8 waves/block
    {
        const int waves  = BB * (NPOINT / 16) * SEGS;  // 2048
        const int blocks = (waves + 7) / 8;            // 256
        qg_ball_kernel<<<blocks, 256, 0, stream>>>(p4, new_xyz, cnts, lists);
    }
    // 2) merge + gather + concat: 8192 waves, 8 waves/block
    {
        const int pairs  = BB * NPOINT;                // 8192
        const int blocks = (pairs + 7) / 8;            // 1024
        qg_group_kernel<<<blocks, 256, 0, stream>>>(xyz, new_xyz, feats,
                                                    cnts, lists, out);
    }
}